// Memory_33483565039939
// MI455X (gfx1250) — compile-verified
//
#include <hip/hip_runtime.h>

#define U 64
#define MSLOTS 8192
#define DDIM 64
#define NEG_INF (-1e9f)

typedef float v2f __attribute__((ext_vector_type(2)));
typedef float v4f __attribute__((ext_vector_type(4)));
typedef float v8f __attribute__((ext_vector_type(8)));

// ---------------------------------------------------------------------------
// K1: scores w[u,m] = dot(attentions[u,m,:], attention[u,:]) via
//     V_WMMA_F32_16X16X4_F32 (full fp32 precision), then mask + temperature.
// grid (8, 64), block 256 (8 waves). Each wave: 128 slots = 8 tiles of 16.
// Two independent WMMA accumulator chains (even/odd K-chunks) for ILP;
// B-matrix columns all replicate the attention chunk so every column of D
// carries the score.
// ---------------------------------------------------------------------------
__global__ __launch_bounds__(256)
void scores_kernel(const float* __restrict__ att,          // [U,D]
                   const float* __restrict__ atts,         // [U,M,D]
                   const float* __restrict__ tmpr,         // [U,1]
                   const unsigned char* __restrict__ mask, // [U,M] (bool)
                   float* __restrict__ wout)               // [U,M]
{
    __shared__ float s_att[DDIM];
    const int u   = blockIdx.y;
    const int tid = threadIdx.x;
    if (tid < DDIM) s_att[tid] = att[u * DDIM + tid];
    __syncthreads();

    const int wave = tid >> 5;
    const int lane = tid & 31;
    const int row  = lane & 15;   // M index within 16-slot tile
    const int hi   = lane >> 4;   // selects K {0,1} vs {2,3} half
    const float t  = tmpr[u];

    const int slot_base = blockIdx.x * 1024 + wave * 128;

    for (int g = 0; g < 8; ++g) {
        const int s0 = slot_base + g * 16;
        const float* abase = atts + ((size_t)u * MSLOTS + s0) * DDIM;

        // gfx1250 prefetch of next tile's row for this lane (near scope)
        __builtin_prefetch(abase + 16 * DDIM + row * DDIM, 0, 3);

        v8f acc0 = {0.f, 0.f, 0.f, 0.f, 0.f, 0.f, 0.f, 0.f};
        v8f acc1 = {0.f, 0.f, 0.f, 0.f, 0.f, 0.f, 0.f, 0.f};
        #pragma unroll
        for (int c = 0; c < 16; c += 2) {
            // A (16x4 f32): vgpr0/1 = K{0,1} for lanes 0-15, K{2,3} for 16-31
            const float* p0 = abase + row * DDIM + 4 * c + 2 * hi;
            v2f A0 = *(const v2f*)p0;
            v2f A1 = *(const v2f*)(p0 + 4);
            // B (4x16 f32): rows replicate the attention chunk across all N
            v2f B0, B1;
            B0.x = s_att[4 * c + 2 * hi];
            B0.y = s_att[4 * c + 2 * hi + 1];
            B1.x = s_att[4 * (c + 1) + 2 * hi];
            B1.y = s_att[4 * (c + 1) + 2 * hi + 1];
            acc0 = __builtin_amdgcn_wmma_f32_16x16x4_f32(
                false, A0, false, B0, (short)0, acc0, false, false);
            acc1 = __builtin_amdgcn_wmma_f32_16x16x4_f32(
                false, A1, false, B1, (short)0, acc1, false, false);
        }
        v8f acc = acc0 + acc1;
        // D layout: lanes 0-15 vgpr r = D[r, lane]; lanes 16-31 = D[8+r, .]
        // all columns identical -> lane 0 writes slots s0..+7, lane 16 rest
        if (row == 0) {
            const int mb = s0 + 8 * hi;
            #pragma unroll
            for (int r = 0; r < 8; ++r) {
                const int m = mb + r;
                float wv = mask[(size_t)u * MSLOTS + m] ? NEG_INF : acc[r];
                wout[(size_t)u * MSLOTS + m] = wv / t;
            }
        }
    }
}

// ---------------------------------------------------------------------------
// K2: per-unit softmax over 8192 slots. 64 blocks x 1024 threads, in-place.
// Each thread owns 8 contiguous floats -> two b128 accesses each way.
// ---------------------------------------------------------------------------
__global__ __launch_bounds__(1024)
void softmax_kernel(float* __restrict__ w) // [U,M] in/out
{
    __shared__ float red[1024];
    const int u   = blockIdx.x;
    const int tid = threadIdx.x;
    float* wu = w + (size_t)u * MSLOTS + (size_t)tid * 8;

    v4f a = *(const v4f*)wu;
    v4f b = *(const v4f*)(wu + 4);

    float mx = fmaxf(fmaxf(fmaxf(a.x, a.y), fmaxf(a.z, a.w)),
                     fmaxf(fmaxf(b.x, b.y), fmaxf(b.z, b.w)));
    red[tid] = mx;
    __syncthreads();
    for (int s = 512; s > 0; s >>= 1) {
        if (tid < s) red[tid] = fmaxf(red[tid], red[tid + s]);
        __syncthreads();
    }
    mx = red[0];
    __syncthreads();

    a.x = __expf(a.x - mx); a.y = __expf(a.y - mx);
    a.z = __expf(a.z - mx); a.w = __expf(a.w - mx);
    b.x = __expf(b.x - mx); b.y = __expf(b.y - mx);
    b.z = __expf(b.z - mx); b.w = __expf(b.w - mx);
    float sum = ((a.x + a.y) + (a.z + a.w)) + ((b.x + b.y) + (b.z + b.w));
    red[tid] = sum;
    __syncthreads();
    for (int s = 512; s > 0; s >>= 1) {
        if (tid < s) red[tid] += red[tid + s];
        __syncthreads();
    }
    const float inv = 1.0f / red[0];
    a *= inv;
    b *= inv;
    *(v4f*)wu       = a;
    *(v4f*)(wu + 4) = b;
}

// ---------------------------------------------------------------------------
// K3: stream memories ONCE: write the output copy and accumulate weighted
// partial outputs. grid (16, 64) = 1024 blocks (full-chip BW), block 256.
// Each 16-lane half-wave owns one slot's full 64-dim row (lane: 4 dims via
// b128); two slots per wave-load -> 512 B fully coalesced per instruction.
// Partials [U,16,D] -> d_ws (256 KB).
// ---------------------------------------------------------------------------
__global__ __launch_bounds__(256)
void outcopy_kernel(const float* __restrict__ mem,    // [U,M,D]
                    const float* __restrict__ wgt,    // [U,M] final weights
                    float* __restrict__ memout,       // [U,M,D]
                    float* __restrict__ partial)      // [U,16,D]
{
    __shared__ float s_part[16][DDIM];
    const int u    = blockIdx.y;
    const int slab = blockIdx.x;          // 0..15
    const int tid  = threadIdx.x;
    const int wave = tid >> 5;
    const int lane = tid & 31;
    const int sub  = lane & 15;           // dim group: dims 4*sub..4*sub+3
    const int sh   = lane >> 4;           // slot parity within pair
    const int m0   = slab * 512 + wave * 64;

    const float* mbase = mem    + ((size_t)u * MSLOTS + m0 + sh) * DDIM + 4 * sub;
    float*       obase = memout + ((size_t)u * MSLOTS + m0 + sh) * DDIM + 4 * sub;
    const float* wb    = wgt    + (size_t)u * MSLOTS + m0 + sh;

    v4f acc = {0.f, 0.f, 0.f, 0.f};
    #pragma unroll 4
    for (int i = 0; i < 32; ++i) {
        v4f v = *(const v4f*)(mbase + (size_t)i * 2 * DDIM);
        *(v4f*)(obase + (size_t)i * 2 * DDIM) = v;
        const float wv = wb[i * 2];
        acc.x = fmaf(wv, v.x, acc.x);
        acc.y = fmaf(wv, v.y, acc.y);
        acc.z = fmaf(wv, v.z, acc.z);
        acc.w = fmaf(wv, v.w, acc.w);
    }
    const int h = wave * 2 + sh;          // 16 half-wave partials
    *(v4f*)&s_part[h][4 * sub] = acc;
    __syncthreads();
    if (tid < DDIM) {
        float s = 0.f;
        #pragma unroll
        for (int k = 0; k < 16; ++k) s += s_part[k][tid];
        partial[((size_t)u * 16 + slab) * DDIM + tid] = s;
    }
}

// ---------------------------------------------------------------------------
// K4: deterministic reduction of 16 slab-partials -> outputs [U,D].
// ---------------------------------------------------------------------------
__global__ __launch_bounds__(64)
void finalize_kernel(const float* __restrict__ partial, // [U,16,D]
                     float* __restrict__ outv)          // [U,D]
{
    const int u = blockIdx.x, d = threadIdx.x;
    float s = 0.f;
    #pragma unroll
    for (int k = 0; k < 16; ++k)
        s += partial[((size_t)u * 16 + k) * DDIM + d];
    outv[u * DDIM + d] = s;
}

// ---------------------------------------------------------------------------
extern "C" void kernel_launch(void* const* d_in, const int* in_sizes, int n_in,
                              void* d_out, int out_size, void* d_ws, size_t ws_size,
                              hipStream_t stream)
{
    const float*         att  = (const float*)d_in[0];         // [64,64]
    const float*         atts = (const float*)d_in[1];         // [64,8192,64]
    const float*         mem  = (const float*)d_in[2];         // [64,8192,64]
    const float*         tmpr = (const float*)d_in[3];         // [64,1]
    const unsigned char* mask = (const unsigned char*)d_in[4]; // [64,8192] bool

    float* out_outputs = (float*)d_out;                        // [64,64]
    float* out_weights = out_outputs + U * DDIM;               // [64,8192]
    float* out_mem     = out_weights + (size_t)U * MSLOTS;     // [64,8192,64]
    float* partial     = (float*)d_ws;                         // [64,16,64] = 256 KB

    scores_kernel  <<<dim3(8,  U), 256,  0, stream>>>(att, atts, tmpr, mask, out_weights);
    softmax_kernel <<<dim3(U),    1024, 0, stream>>>(out_weights);
    outcopy_kernel <<<dim3(16, U), 256,  0, stream>>>(mem, out_weights, out_mem, partial);
    finalize_kernel<<<dim3(U),     64,   0, stream>>>(partial, out_outputs);
}